// GNN_69690139345247
// MI455X (gfx1250) — compile-verified
//
#include <hip/hip_runtime.h>
#include <math.h>

// Problem constants (match reference)
#define NN   25000   // nodes
#define EE   100000  // edges
#define FIN  16      // input features
#define NH   4       // heads
#define CC   280     // channels per head
#define HC   1120    // NH*CC
#define EDIM 6       // edge feature dim
#define NOUT 6       // output classes

typedef __attribute__((ext_vector_type(2))) float v2f;
typedef __attribute__((ext_vector_type(8))) float v8f;

static __device__ __forceinline__ v8f wmma4(v2f a, v2f b, v8f c) {
  // V_WMMA_F32_16X16X4_F32 : full-fp32 matrix op, D = A(16x4) * B(4x16) + C
  return __builtin_amdgcn_wmma_f32_16x16x4_f32(
      /*neg_a=*/false, a, /*neg_b=*/false, b,
      /*c_mod=*/(short)0, c, /*reuse_a=*/false, /*reuse_b=*/false);
}

// Order-preserving float <-> uint mapping for atomicMax on floats
static __device__ __forceinline__ unsigned f2ord(float f) {
  unsigned u = __float_as_uint(f);
  return (u & 0x80000000u) ? ~u : (u | 0x80000000u);
}
static __device__ __forceinline__ float ord2f(unsigned u) {
  return (u & 0x80000000u) ? __uint_as_float(u & 0x7FFFFFFFu)
                           : __uint_as_float(~u);
}

// ---------------------------------------------------------------------------
// Init: zero the attention accumulator and the segment max/sum buffers.
// Must run every launch (harness poisons ws once, never restores).
// ---------------------------------------------------------------------------
__global__ void init_kernel(float* __restrict__ oacc,
                            unsigned* __restrict__ amax,
                            float* __restrict__ asum) {
  int i = blockIdx.x * blockDim.x + threadIdx.x;
  if (i < NN * CC) oacc[i] = 0.0f;
  if (i < NN * NH) { amax[i] = 0u; asum[i] = 0.0f; }
}

// ---------------------------------------------------------------------------
// Generic fp32 WMMA GEMM:  D = act( A[M x KD] * W[KD x ND] + bias (+ D) )
// One wave computes one 16x16 tile of D using KD/4 chained v_wmma_f32_16x16x4.
// KD must be a multiple of 4 (16, 280, 64 all are).
// ---------------------------------------------------------------------------
template <int KD, int ND, bool RELU, bool ACCUM>
__global__ void gemm_bias_kernel(const float* __restrict__ A,
                                 const float* __restrict__ W,
                                 const float* __restrict__ bias,
                                 float* __restrict__ D, int M) {
  const int NT = (ND + 15) / 16;
  const int MT = (M + 15) / 16;
  int wave = blockIdx.x * (blockDim.x >> 5) + (threadIdx.x >> 5);
  if (wave >= MT * NT) return;                 // wave-uniform: EXEC stays full
  const int mt = wave / NT, nt = wave % NT;
  const int lane = threadIdx.x & 31;
  const int lm = lane & 15, lh = lane >> 4;
  const int m0 = mt * 16, n0 = nt * 16;

  int mload = m0 + lm; if (mload >= M) mload = M - 1;       // clamp (masked on store)
  const int nc = n0 + lm;
  const int nload = (nc < ND) ? nc : (ND - 1);              // clamp (masked on store)

  const float* arow = A + (size_t)mload * KD;
  v8f c = {0.f, 0.f, 0.f, 0.f, 0.f, 0.f, 0.f, 0.f};

#pragma unroll 4
  for (int s = 0; s < KD / 4; ++s) {
    const int kb = 4 * s + 2 * lh;            // ISA 16x4 A layout: K = 2*(lane/16)+v
    v2f a, b;
    a.x = arow[kb];
    a.y = arow[kb + 1];
    b.x = W[(size_t)kb * ND + nload];          // B 4x16: row striped across lanes
    b.y = W[(size_t)(kb + 1) * ND + nload];
    c = wmma4(a, b, c);
  }

  if (nc < ND) {
    const float bn = bias[nc];
#pragma unroll
    for (int r = 0; r < 8; ++r) {             // C/D: VGPR r -> rows r and r+8
      const int mm = m0 + r + 8 * lh;
      if (mm < M) {
        float val = c[r] + bn;
        const size_t idx = (size_t)mm * ND + nc;
        if (ACCUM) val += D[idx];
        if (RELU)  val = fmaxf(val, 0.0f);
        D[idx] = val;
      }
    }
  }
}

// ---------------------------------------------------------------------------
// Pass A: one wave per (edge, head).
// alpha[e,h] = <q[dst,h,:], k[src,h,:] + (edge_attr[e] @ We)[h,:]> / sqrt(C)
// and segment-max into amax[dst,h] (ordered-uint atomicMax).
// ---------------------------------------------------------------------------
__global__ void edge_alpha_kernel(const int* __restrict__ ei,
                                  const float* __restrict__ ea,
                                  const float* __restrict__ We,
                                  const float* __restrict__ q,
                                  const float* __restrict__ k,
                                  float* __restrict__ alpha,
                                  unsigned* __restrict__ amax) {
  int wave = blockIdx.x * (blockDim.x >> 5) + (threadIdx.x >> 5);
  if (wave >= EE * NH) return;
  const int e = wave >> 2, h = wave & 3;
  const int lane = threadIdx.x & 31;
  const int src = ei[e], dst = ei[EE + e];

  const float w0 = ea[e * EDIM + 0], w1 = ea[e * EDIM + 1], w2 = ea[e * EDIM + 2];
  const float w3 = ea[e * EDIM + 3], w4 = ea[e * EDIM + 4], w5 = ea[e * EDIM + 5];

  const float* qrow = q + (size_t)dst * HC + h * CC;
  const float* krow = k + (size_t)src * HC + h * CC;
  const float* Web  = We + h * CC;

  float acc = 0.0f;
  for (int c = lane; c < CC; c += 32) {
    const float ec = w0 * Web[c]          + w1 * Web[HC + c]     + w2 * Web[2 * HC + c]
                   + w3 * Web[3 * HC + c] + w4 * Web[4 * HC + c] + w5 * Web[5 * HC + c];
    acc += qrow[c] * (krow[c] + ec);
  }
  for (int off = 16; off > 0; off >>= 1) acc += __shfl_xor(acc, off);

  if (lane == 0) {
    const float a = acc * 0.05976143046671968f;   // 1/sqrt(280)
    alpha[wave] = a;
    atomicMax(&amax[dst * NH + h], f2ord(a));
  }
}

// ---------------------------------------------------------------------------
// Pass B: ex = exp(alpha - amax[dst]); segment-sum into asum[dst,h].
// ---------------------------------------------------------------------------
__global__ void edge_exp_kernel(const int* __restrict__ ei,
                                float* __restrict__ alpha,
                                const unsigned* __restrict__ amax,
                                float* __restrict__ asum) {
  int i = blockIdx.x * blockDim.x + threadIdx.x;
  if (i >= EE * NH) return;
  const int e = i >> 2, h = i & 3;
  const int dst = ei[EE + e];
  const float mx = ord2f(amax[dst * NH + h]);
  const float ex = expf(alpha[i] - mx);
  alpha[i] = ex;                                    // overwrite in place
  atomicAdd(&asum[dst * NH + h], ex);
}

// ---------------------------------------------------------------------------
// Pass C: one wave per (edge, head).
// out[dst, c] += (1/H) * attn * (v[src,h,c] + e[h,c])   (head-mean folded in)
// ---------------------------------------------------------------------------
__global__ void edge_accum_kernel(const int* __restrict__ ei,
                                  const float* __restrict__ ea,
                                  const float* __restrict__ We,
                                  const float* __restrict__ v,
                                  const float* __restrict__ alpha,
                                  const float* __restrict__ asum,
                                  float* __restrict__ out) {
  int wave = blockIdx.x * (blockDim.x >> 5) + (threadIdx.x >> 5);
  if (wave >= EE * NH) return;
  const int e = wave >> 2, h = wave & 3;
  const int lane = threadIdx.x & 31;
  const int src = ei[e], dst = ei[EE + e];

  const float attn = alpha[wave] / (asum[dst * NH + h] + 1e-16f);
  const float scale = attn * 0.25f;                 // mean over H=4 heads

  const float w0 = ea[e * EDIM + 0], w1 = ea[e * EDIM + 1], w2 = ea[e * EDIM + 2];
  const float w3 = ea[e * EDIM + 3], w4 = ea[e * EDIM + 4], w5 = ea[e * EDIM + 5];

  const float* vrow = v + (size_t)src * HC + h * CC;
  const float* Web  = We + h * CC;
  float* orow = out + (size_t)dst * CC;

  for (int c = lane; c < CC; c += 32) {
    const float ec = w0 * Web[c]          + w1 * Web[HC + c]     + w2 * Web[2 * HC + c]
                   + w3 * Web[3 * HC + c] + w4 * Web[4 * HC + c] + w5 * Web[5 * HC + c];
    atomicAdd(&orow[c], scale * (vrow[c] + ec));
  }
}

// ---------------------------------------------------------------------------
// Final layer: logits = h2 @ W3 + b3 (K=16, N=6), row softmax. One thread/node.
// ---------------------------------------------------------------------------
__global__ void final_softmax_kernel(const float* __restrict__ h2,
                                     const float* __restrict__ W3,
                                     const float* __restrict__ b3,
                                     float* __restrict__ out) {
  int n = blockIdx.x * blockDim.x + threadIdx.x;
  if (n >= NN) return;
  float hi[16];
#pragma unroll
  for (int i = 0; i < 16; ++i) hi[i] = h2[(size_t)n * 16 + i];
  float l[NOUT];
#pragma unroll
  for (int j = 0; j < NOUT; ++j) {
    float s = b3[j];
#pragma unroll
    for (int i = 0; i < 16; ++i) s += hi[i] * W3[i * NOUT + j];
    l[j] = s;
  }
  float m = l[0];
#pragma unroll
  for (int j = 1; j < NOUT; ++j) m = fmaxf(m, l[j]);
  float sum = 0.0f;
#pragma unroll
  for (int j = 0; j < NOUT; ++j) { l[j] = expf(l[j] - m); sum += l[j]; }
  const float inv = 1.0f / sum;
#pragma unroll
  for (int j = 0; j < NOUT; ++j) out[(size_t)n * NOUT + j] = l[j] * inv;
}

// ---------------------------------------------------------------------------
// Host launcher
// ---------------------------------------------------------------------------
extern "C" void kernel_launch(void* const* d_in, const int* in_sizes, int n_in,
                              void* d_out, int out_size, void* d_ws, size_t ws_size,
                              hipStream_t stream) {
  const float* x     = (const float*)d_in[0];
  const int*   ei    = (const int*)d_in[1];
  const float* ea    = (const float*)d_in[2];
  const float* Wq    = (const float*)d_in[3];
  const float* bq    = (const float*)d_in[4];
  const float* Wk    = (const float*)d_in[5];
  const float* bk    = (const float*)d_in[6];
  const float* Wv    = (const float*)d_in[7];
  const float* bv    = (const float*)d_in[8];
  const float* We    = (const float*)d_in[9];
  const float* Wskip = (const float*)d_in[10];
  const float* bskip = (const float*)d_in[11];
  const float* W1    = (const float*)d_in[12];
  const float* b1    = (const float*)d_in[13];
  const float* W2    = (const float*)d_in[14];
  const float* b2    = (const float*)d_in[15];
  const float* W3    = (const float*)d_in[16];
  const float* b3    = (const float*)d_in[17];
  float* out = (float*)d_out;

  // Workspace layout (~375 MB)
  float* q     = (float*)d_ws;                    // N*HC
  float* k     = q + (size_t)NN * HC;             // N*HC
  float* v     = k + (size_t)NN * HC;             // N*HC
  float* oacc  = v + (size_t)NN * HC;             // N*CC  (attention output, head-mean)
  float* alpha = oacc + (size_t)NN * CC;          // E*NH
  unsigned* amax = (unsigned*)(alpha + (size_t)EE * NH);  // N*NH
  float* asum  = (float*)(amax + (size_t)NN * NH);        // N*NH
  float* h1    = asum + (size_t)NN * NH;          // N*64
  float* h2    = h1 + (size_t)NN * 64;            // N*16

  const dim3 blk(256);
  auto tiles = [](int M, int Nd) { return ((M + 15) / 16) * ((Nd + 15) / 16); };

  // 0) re-init accumulators every call
  init_kernel<<<(NN * CC + 255) / 256, blk, 0, stream>>>(oacc, amax, asum);

  // 1) q,k,v = x @ W{q,k,v} + b   (fp32 WMMA)
  const int tQKV = tiles(NN, HC);
  gemm_bias_kernel<FIN, HC, false, false><<<(tQKV + 7) / 8, blk, 0, stream>>>(x, Wq, bq, q, NN);
  gemm_bias_kernel<FIN, HC, false, false><<<(tQKV + 7) / 8, blk, 0, stream>>>(x, Wk, bk, k, NN);
  gemm_bias_kernel<FIN, HC, false, false><<<(tQKV + 7) / 8, blk, 0, stream>>>(x, Wv, bv, v, NN);

  // 2) attention: alpha + segment max, exp + segment sum, weighted scatter
  const int ewaves = EE * NH;
  edge_alpha_kernel<<<(ewaves + 7) / 8, blk, 0, stream>>>(ei, ea, We, q, k, alpha, amax);
  edge_exp_kernel<<<(ewaves + 255) / 256, blk, 0, stream>>>(ei, alpha, amax, asum);
  edge_accum_kernel<<<(ewaves + 7) / 8, blk, 0, stream>>>(ei, ea, We, v, alpha, asum, oacc);

  // 3) skip connection in place: oacc += x @ Wskip + bskip   (fp32 WMMA)
  gemm_bias_kernel<FIN, CC, false, true><<<(tiles(NN, CC) + 7) / 8, blk, 0, stream>>>(
      x, Wskip, bskip, oacc, NN);

  // 4) MLP: relu(oacc @ W1 + b1) -> relu(h1 @ W2 + b2)   (fp32 WMMA)
  gemm_bias_kernel<CC, 64, true, false><<<(tiles(NN, 64) + 7) / 8, blk, 0, stream>>>(
      oacc, W1, b1, h1, NN);
  gemm_bias_kernel<64, 16, true, false><<<(tiles(NN, 16) + 7) / 8, blk, 0, stream>>>(
      h1, W2, b2, h2, NN);

  // 5) final logits + softmax
  final_softmax_kernel<<<(NN + 255) / 256, blk, 0, stream>>>(h2, W3, b3, out);
}